// BilateralGrid_51677046506219
// MI455X (gfx1250) — compile-verified
//
#include <hip/hip_runtime.h>
#include <stdint.h>

// Problem constants (fixed by the reference setup_inputs()).
#define NVIEW 2
#define LDEP 8
#define GH 16
#define GW 16
#define IMH 1080
#define IMW 1920
#define PIX_PER_VIEW (IMH * IMW)              // 2,073,600
#define GROUPS_PER_VIEW (PIX_PER_VIEW / 4)    // 518,400 (4 pixels per group)
#define THREADS 256
#define ITERS 4
#define GROUPS_PER_BLOCK (THREADS * ITERS)    // 1024
#define BLOCKS_PER_VIEW ((GROUPS_PER_VIEW + GROUPS_PER_BLOCK - 1) / GROUPS_PER_BLOCK) // 507
#define GRID_ELEMS (12 * LDEP * GH * GW)      // 24,576 floats per view (96 KB)

// gfx1250 async global->LDS copy path (probe-confirmed present; builtin takes
// address-space-qualified int* operands per the compiler diagnostic).
#if __has_builtin(__builtin_amdgcn_global_load_async_to_lds_b32) && \
    __has_builtin(__builtin_amdgcn_s_wait_asynccnt)
#define USE_ASYNC_LDS 1
#endif

typedef __attribute__((address_space(1))) int* as1_ip;  // global
typedef __attribute__((address_space(3))) int* as3_ip;  // LDS

union F4 {
    float4 v;
    float f[4];
};

__global__ __launch_bounds__(THREADS) void bgrid_slice_kernel(
    const float* __restrict__ grids,   // (N,12,L,H,W)
    const float* __restrict__ coords,  // (N,1,h,w,2)
    const float* __restrict__ rgb,     // (N,1,h,w,3)
    float* __restrict__ out)           // (N,1,h,w,3)
{
    // LDS grid, transposed to (z,y,x,c) so one corner = 12 contiguous floats
    // (48 B, 16B-aligned -> 3x ds_load_b128). 96 KB: only possible on CDNA5's
    // 320 KB-per-WGP LDS.
    __shared__ __align__(16) float sg[GRID_ELEMS];

    const int n   = blockIdx.y;
    const int tid = threadIdx.x;
    const float* gsrc = grids + (size_t)n * GRID_ELEMS;

    // ---- Stage grid into LDS (transpose (c, s) -> (s, c); s = z*256+y*16+x) ----
#ifdef USE_ASYNC_LDS
    #pragma unroll 1
    for (int i = tid; i < GRID_ELEMS; i += THREADS) {
        const int c = i >> 11;     // i / 2048
        const int s = i & 2047;    // i % 2048
        as1_ip gp = (as1_ip)(uintptr_t)(gsrc + i);
        as3_ip lp = (as3_ip)(uintptr_t)(&sg[s * 12 + c]);
        __builtin_amdgcn_global_load_async_to_lds_b32(gp, lp, 0, 0);
    }
    __builtin_amdgcn_s_wait_asynccnt(0);
    __syncthreads();
#else
    #pragma unroll 1
    for (int i = tid; i < GRID_ELEMS; i += THREADS) {
        const int c = i >> 11;
        const int s = i & 2047;
        sg[s * 12 + c] = gsrc[i];
    }
    __syncthreads();
#endif

    const long long viewPixBase  = (long long)n * PIX_PER_VIEW;
    const int blockGroupBase     = blockIdx.x * GROUPS_PER_BLOCK;

    #pragma unroll 1
    for (int it = 0; it < ITERS; ++it) {
        const int grp = blockGroupBase + it * THREADS + tid;
        if (grp >= GROUPS_PER_VIEW) break;
        const long long p = viewPixBase + (long long)grp * 4;  // base pixel of group

        // Prefetch next iteration's streams (gfx1250 global_prefetch_b8).
        const int ngrp = grp + THREADS;
        if ((it + 1 < ITERS) && (ngrp < GROUPS_PER_VIEW)) {
            const long long np = viewPixBase + (long long)ngrp * 4;
            __builtin_prefetch(rgb + np * 3, 0, 1);
            __builtin_prefetch(coords + np * 2, 0, 1);
        }

        // 4 pixels per thread -> all global traffic is b128.
        const float4* cvec = (const float4*)(coords + p * 2);
        const float4  c0 = cvec[0], c1 = cvec[1];
        const float4* rvec = (const float4*)(rgb + p * 3);
        const float4  q0 = rvec[0], q1 = rvec[1], q2 = rvec[2];

        const float cxs[4] = {c0.x, c0.z, c1.x, c1.z};
        const float cys[4] = {c0.y, c0.w, c1.y, c1.w};
        const float rs[4]  = {q0.x, q0.w, q1.z, q2.y};
        const float gs[4]  = {q0.y, q1.x, q1.w, q2.z};
        const float bs[4]  = {q0.z, q1.y, q2.x, q2.w};

        float o[12];
        #pragma unroll
        for (int j = 0; j < 4; ++j) {
            const float r = rs[j], g = gs[j], b = bs[j];
            const float gray = fmaf(0.299f, r, fmaf(0.587f, g, 0.114f * b));

            const float x = cxs[j] * (float)(GW - 1);
            const float y = cys[j] * (float)(GH - 1);
            const float z = gray   * (float)(LDEP - 1);

            float fx0 = fminf(fmaxf(floorf(x), 0.f), (float)(GW - 2));
            float fy0 = fminf(fmaxf(floorf(y), 0.f), (float)(GH - 2));
            float fz0 = fminf(fmaxf(floorf(z), 0.f), (float)(LDEP - 2));
            const int x0 = (int)fx0, y0 = (int)fy0, z0 = (int)fz0;
            const float fx = x - fx0, fy = y - fy0, fz = z - fz0;

            float A[12];
            #pragma unroll
            for (int k = 0; k < 12; ++k) A[k] = 0.f;

            const int base = (z0 * (GH * GW) + y0 * GW + x0) * 12;
            #pragma unroll
            for (int dz = 0; dz < 2; ++dz) {
                const float wz = dz ? fz : (1.f - fz);
                #pragma unroll
                for (int dy = 0; dy < 2; ++dy) {
                    const float wy = dy ? fy : (1.f - fy);
                    #pragma unroll
                    for (int dx = 0; dx < 2; ++dx) {
                        const float wx  = dx ? fx : (1.f - fx);
                        const float wgt = wz * wy * wx;
                        const int off = base + (dz * (GH * GW) + dy * GW + dx) * 12;
                        const float4* cp = (const float4*)(sg + off);  // 16B aligned
                        F4 a0, a1, a2;
                        a0.v = cp[0]; a1.v = cp[1]; a2.v = cp[2];
                        #pragma unroll
                        for (int k = 0; k < 4; ++k) {
                            A[k]     = fmaf(wgt, a0.f[k], A[k]);
                            A[4 + k] = fmaf(wgt, a1.f[k], A[4 + k]);
                            A[8 + k] = fmaf(wgt, a2.f[k], A[8 + k]);
                        }
                    }
                }
            }

            // out = A[:, :3] @ rgb + A[:, 3]
            o[j * 3 + 0] = fmaf(A[0], r, fmaf(A[1],  g, fmaf(A[2],  b, A[3])));
            o[j * 3 + 1] = fmaf(A[4], r, fmaf(A[5],  g, fmaf(A[6],  b, A[7])));
            o[j * 3 + 2] = fmaf(A[8], r, fmaf(A[9],  g, fmaf(A[10], b, A[11])));
        }

        float4* ovec = (float4*)(out + p * 3);
        ovec[0] = make_float4(o[0], o[1], o[2],  o[3]);
        ovec[1] = make_float4(o[4], o[5], o[6],  o[7]);
        ovec[2] = make_float4(o[8], o[9], o[10], o[11]);
    }
}

extern "C" void kernel_launch(void* const* d_in, const int* in_sizes, int n_in,
                              void* d_out, int out_size, void* d_ws, size_t ws_size,
                              hipStream_t stream) {
    (void)in_sizes; (void)n_in; (void)out_size; (void)d_ws; (void)ws_size;
    const float* grids  = (const float*)d_in[0];
    const float* coords = (const float*)d_in[1];
    const float* rgb    = (const float*)d_in[2];
    float* out = (float*)d_out;

    dim3 grid(BLOCKS_PER_VIEW, NVIEW, 1);
    dim3 block(THREADS, 1, 1);
    bgrid_slice_kernel<<<grid, block, 0, stream>>>(grids, coords, rgb, out);
}